// QPLayer_16492674417435
// MI455X (gfx1250) — compile-verified
//
#include <hip/hip_runtime.h>
#include <hip/hip_bf16.h>

// Batched QP (FISTA on the dual), one workgroup per batch instance.
//   G = A D A^T = R R^T  (R = A*sqrt(D), D = inv(diag(Q)))  -> f32 WMMA GEMM
//   iterate 200x:  g = c - G y ;  lam = relu(y + eta g) ; momentum
// G lives entirely in VGPRs (WMMA accumulators), exploiting G's symmetry so
// each lane owns one G row. y is double-buffered in LDS (broadcast reads).

typedef __attribute__((ext_vector_type(2))) float v2f;
typedef __attribute__((ext_vector_type(8))) float v8f;

#define QP_ITERS   200
#define QP_M       128
#define QP_N       64
#define RSTRIDE    66          // even padding: aligned v2f loads, decent banks
#define NTHREADS   256         // 8 wave32

__global__ __launch_bounds__(NTHREADS) void qp_fista_kernel(
    const float* __restrict__ Qm, const float* __restrict__ A,
    const float* __restrict__ X,  const float* __restrict__ Bv,
    float* __restrict__ Out)
{
    __shared__ __align__(16) float sR[QP_M * RSTRIDE];   // R = A * sqrt(dinv)
    __shared__ __align__(16) float sY[2][QP_M];          // double-buffered y
    __shared__ float sC[QP_M];
    __shared__ float sLam[QP_M];
    __shared__ float sDinv[QP_N];
    __shared__ float sSqrtd[QP_N];
    __shared__ float sXp[QP_N];                          // sqrt(dinv)*x
    __shared__ float sRed[8];
    __shared__ float sEta;

    const int tid   = threadIdx.x;
    const int lane  = tid & 31;
    const int wave  = tid >> 5;
    const int ln16  = lane & 15;
    const int hl    = lane >> 4;         // 0: cols +0..7, 1: cols +8..15
    const int batch = blockIdx.x;

    const float* Ab = A  + (size_t)batch * (QP_M * QP_N);
    const float* xb = X  + (size_t)batch * QP_N;
    const float* bb = Bv + (size_t)batch * QP_M;

    // ---- Phase 1: diag(Q) inverse ----------------------------------------
    if (tid < QP_N) {
        float di = 1.0f / Qm[tid * QP_N + tid];
        sDinv[tid]  = di;
        sSqrtd[tid] = sqrtf(fabsf(di));
    }
    __syncthreads();
    if (tid < QP_N) sXp[tid] = sSqrtd[tid] * xb[tid];

    // ---- Phase 2: load A -> R in LDS, accumulate sum(A^2) ----------------
    float ss = 0.f;
    #pragma unroll
    for (int j = 0; j < (QP_M * QP_N / 4) / NTHREADS; ++j) {   // 2048 f4 / 256
        int f4  = j * NTHREADS + tid;
        float4 v = ((const float4*)Ab)[f4];
        int fl  = f4 << 2;
        int row = fl >> 6, col = fl & 63;
        ss += v.x*v.x + v.y*v.y + v.z*v.z + v.w*v.w;
        float* dst = &sR[row * RSTRIDE + col];
        dst[0] = v.x * sSqrtd[col + 0];
        dst[1] = v.y * sSqrtd[col + 1];
        dst[2] = v.z * sSqrtd[col + 2];
        dst[3] = v.w * sSqrtd[col + 3];
    }
    #pragma unroll
    for (int off = 16; off; off >>= 1) ss += __shfl_xor(ss, off, 32);
    if (lane == 0) sRed[wave] = ss;
    if (tid < QP_M) sY[0][tid] = 0.f;
    __syncthreads();

    if (tid == 0) {
        float s = 0.f;
        for (int w = 0; w < 8; ++w) s += sRed[w];
        float qn2 = 0.f;
        for (int n = 0; n < QP_N; ++n) qn2 += sDinv[n] * sDinv[n];
        float L = s * sqrtf(qn2);                  // ||A||_F^2 * ||Qinv||_F
        sEta = 1.0f / fmaxf(L, 1e-12f);
    }
    // c = R * (sqrt(dinv)*x) - b
    if (tid < QP_M) {
        float acc = 0.f;
        #pragma unroll 8
        for (int n = 0; n < QP_N; ++n) acc += sR[tid * RSTRIDE + n] * sXp[n];
        sC[tid] = acc - bb[tid];
    }
    __syncthreads();

    // ---- Phase 3: G = R R^T via V_WMMA_F32_16X16X4_F32 -------------------
    // Wave w computes tiles (i, j=w), i=0..7. By symmetry of G, lane l then
    // owns row r = 16w + (l&15), columns 16i + 8*hl + v (v=0..7) in acc[i][v].
    v2f bf[16];                                   // B-frags (j = wave), cached
    #pragma unroll
    for (int k = 0; k < 16; ++k) {
        int row = wave * 16 + ln16;
        int col = 4 * k + 2 * hl;                 // even -> 8B aligned
        bf[k] = *(const v2f*)&sR[row * RSTRIDE + col];
    }
    v8f acc[8];
    #pragma unroll
    for (int i = 0; i < 8; ++i) {
        v8f c = {};
        #pragma unroll
        for (int k = 0; k < 16; ++k) {
            int row = i * 16 + ln16;
            int col = 4 * k + 2 * hl;
            v2f af = *(const v2f*)&sR[row * RSTRIDE + col];
            // D = A(16x4 f32) * B(4x16 f32) + C, wave32
            c = __builtin_amdgcn_wmma_f32_16x16x4_f32(
                    false, af, false, bf[k], (short)0, c, false, false);
        }
        acc[i] = c;
    }
    __syncthreads();

    // ---- Phase 4: 200 FISTA iterations, G register-resident ---------------
    const float eta = sEta;
    const int   r   = wave * 16 + ln16;           // G row owned by this lane
    float c_r = sC[r];
    float lam_r = 0.f, y_r = 0.f, t = 1.f;
    int buf = 0;

    for (int it = 0; it < QP_ITERS; ++it) {
        const float* y = sY[buf];
        float a = 0.f;
        #pragma unroll
        for (int i = 0; i < 8; ++i) {
            // broadcast reads: all lanes of a half-wave hit the same address
            float4 y0 = *(const float4*)&y[i * 16 + 8 * hl];
            float4 y1 = *(const float4*)&y[i * 16 + 8 * hl + 4];
            a += acc[i][0]*y0.x + acc[i][1]*y0.y + acc[i][2]*y0.z + acc[i][3]*y0.w
               + acc[i][4]*y1.x + acc[i][5]*y1.y + acc[i][6]*y1.z + acc[i][7]*y1.w;
        }
        a += __shfl_xor(a, 16, 32);               // combine the two row halves
        float tn = 0.5f * (1.f + sqrtf(1.f + 4.f * t * t));
        if (lane < 16) {
            float g   = c_r - a;                  // -(G y - c) = -(A z - b) sign-fixed below
            float ln_ = fmaxf(0.f, y_r + eta * g);
            float yn  = ln_ + ((t - 1.f) / tn) * (ln_ - lam_r);
            sY[buf ^ 1][r] = yn;
            lam_r = ln_;
            y_r   = yn;
        }
        t = tn;
        buf ^= 1;
        __syncthreads();
    }

    // ---- Phase 5: recover z = D x - sqrt(D) * (R^T lam) -------------------
    if (lane < 16) sLam[r] = lam_r;
    __syncthreads();
    if (tid < QP_N) {
        float a2 = 0.f;
        #pragma unroll 8
        for (int m = 0; m < QP_M; ++m) a2 += sR[m * RSTRIDE + tid] * sLam[m];
        Out[(size_t)batch * QP_N + tid] = sDinv[tid] * xb[tid] - sSqrtd[tid] * a2;
    }
}

extern "C" void kernel_launch(void* const* d_in, const int* in_sizes, int n_in,
                              void* d_out, int out_size, void* d_ws, size_t ws_size,
                              hipStream_t stream) {
    const float* Q = (const float*)d_in[0];   // [64,64]
    const float* A = (const float*)d_in[1];   // [B,128,64]
    const float* X = (const float*)d_in[2];   // [B,64]
    const float* B = (const float*)d_in[3];   // [B,128]
    float* Out = (float*)d_out;               // [B,64]
    int batches = in_sizes[1] / (QP_M * QP_N);
    qp_fista_kernel<<<batches, NTHREADS, 0, stream>>>(Q, A, X, B, Out);
}